// VitGnnModel_33217277067878
// MI455X (gfx1250) — compile-verified
//
#include <hip/hip_runtime.h>
#include <hip/hip_bf16.h>
#include <stdint.h>

// ---------------- problem constants ----------------
#define NIMG     32
#define D_ELEMS  196608              // 3*256*256 floats per image
#define CHUNK    256                 // K-chunk (floats per image per TDM tile)
#define NCHUNK   (D_ELEMS / CHUNK)   // 768
#define KBLOCKS  256                 // K partitioned across blocks; 768/256 = 3 chunks/block
#define LDS_STRIDE (CHUNK + 4)       // +4 dword pad per row (TDM pad) -> conflict-free banks
#define BUF_FLOATS (NIMG * LDS_STRIDE)   // 8320 floats per buffer
#define BUF_BYTES  (BUF_FLOATS * 4)      // 33280 B; x2 buffers = 66560 B LDS
#define HQUADS   (CHUNK / 8)         // 32 float4-quads = half of a chunk's K range

typedef unsigned int u32x4 __attribute__((ext_vector_type(4)));
typedef int          i32x8 __attribute__((ext_vector_type(8)));
typedef int          i32x4 __attribute__((ext_vector_type(4)));
typedef float        f32x4 __attribute__((ext_vector_type(4)));

// ---------------- TDM: one 32-row x CHUNK-float strided tile -> LDS ----------------
// D# group0: [1:0]=count=1, [63:32]=lds_addr, [120:64]=global_addr, [127:126]=type=2
// D# group1: data_size=4B, pad_enable, pad_interval=256 dwords, pad_amount=4 dwords,
//            tensor_dim0=CHUNK, tensor_dim1=NIMG, tile_dim0=CHUNK, tile_dim1=NIMG,
//            tensor_dim0_stride=D_ELEMS (row stride between images)
__device__ __forceinline__ void tdm_load_tile(uint32_t lds_byte_addr, const float* gsrc) {
    uint64_t ga = (uint64_t)(uintptr_t)gsrc;
    u32x4 g0;
    g0[0] = 1u;                                    // count = 1 (valid user descriptor)
    g0[1] = lds_byte_addr;                         // LDS destination (bytes)
    g0[2] = (uint32_t)ga;                          // global_addr[31:0]
    g0[3] = (uint32_t)(ga >> 32) | (2u << 30);     // global_addr[56:32] | type=2 ("image")
    i32x8 g1;
    g1[0] = (2 << 16)                              // data_size = 4 bytes
          | (1 << 20)                              // pad_enable
          | (7 << 22)                              // pad_interval: 256 dwords (one row)
          | (3 << 25);                             // pad_amount : 4 dwords  -> stride 260
    g1[1] = (CHUNK & 0xFFFF) << 16;                // tensor_dim0[15:0]
    g1[2] = (NIMG  & 0xFFFF) << 16;                // tensor_dim0 hi=0 | tensor_dim1[15:0]
    g1[3] = (CHUNK << 16);                         // tensor_dim1 hi=0 | tile_dim0
    g1[4] = NIMG;                                  // tile_dim1 | tile_dim2=0
    g1[5] = D_ELEMS;                               // tensor_dim0_stride[31:0] (elem units)
    g1[6] = 0;                                     // stride hi | dim1_stride lo
    g1[7] = 0;
    i32x4 g2; g2[0] = 1; g2[1] = 1; g2[2] = 0; g2[3] = 0;        // tensor_dim2/3 = 1 (unused)
    i32x4 g3; g3[0] = 0; g3[1] = (1 << 16); g3[2] = 0; g3[3] = 0; // tensor_dim4 = 1 (unused)
    i32x8 g4 = {0, 0, 0, 0, 0, 0, 0, 0};                          // extra group (clang-23 form)
    __builtin_amdgcn_tensor_load_to_lds(g0, g1, g2, g3, g4, 0);
}

// ---------------- main kernel: per-K-block partial L1 sums, symmetry-halved ----------
// Thread tile (m0,n0) covers pairs {m0,m0+16} x {n0,n0+16}. Per-lane K-half offsets:
//   s01 -> A0,B1 pointers;  s10 -> A1,B0 pointers.
// Upper-triangle tiles (m0<n0) take K-half 0; lower-triangle (m0>n0) K-half 1; the
// mirror tile supplies the complementary half, summed in finalize. Diagonal tiles get
// the (a,a+16) pair split across acc01 (half 0) / acc10 (half 1); their acc00/acc11
// slots are garbage but correspond to diagonal outputs which are exactly 0.
__global__ __launch_bounds__(256) void pairwise_l1_partials(
    const float* __restrict__ images, float* __restrict__ partials) {
    __shared__ float smem[2 * BUF_FLOATS];

    const int t = threadIdx.x;
    // wave-uniform: true only for wave 0 -> scalar branch guards TDM issue (TDM ignores EXEC)
    const bool issuer = (__builtin_amdgcn_readfirstlane((int)t) == 0);
    const uint32_t lds0 = (uint32_t)(uintptr_t)(&smem[0]);

    const int m0 = t >> 4, m1 = m0 + 16;
    const int n0 = t & 15, n1 = n0 + 16;
    const int s01 = (m0 >  n0) ? HQUADS : 0;   // quad offset for A0 / B1
    const int s10 = (m0 >= n0) ? HQUADS : 0;   // quad offset for A1 / B0

    float acc00 = 0.f, acc01 = 0.f, acc10 = 0.f, acc11 = 0.f;

    int c = blockIdx.x;
    if (issuer) tdm_load_tile(lds0, images + (size_t)c * CHUNK);   // prologue: chunk c -> buf0
    int phase = 0;

    for (; c < NCHUNK; c += KBLOCKS) {
        const int cn = c + KBLOCKS;
        if (issuer) {
            if (cn < NCHUNK) {  // prefetch next chunk into the other buffer
                tdm_load_tile(lds0 + (uint32_t)((phase ^ 1) * BUF_BYTES),
                              images + (size_t)cn * CHUNK);
                __builtin_amdgcn_s_wait_tensorcnt(1);   // current chunk landed (in-order)
            } else {
                __builtin_amdgcn_s_wait_tensorcnt(0);
            }
        }
        __syncthreads();

        const float* buf = smem + phase * BUF_FLOATS;
        const f32x4* A0 = (const f32x4*)(buf + m0 * LDS_STRIDE) + s01;
        const f32x4* A1 = (const f32x4*)(buf + m1 * LDS_STRIDE) + s10;
        const f32x4* B0 = (const f32x4*)(buf + n0 * LDS_STRIDE) + s10;
        const f32x4* B1 = (const f32x4*)(buf + n1 * LDS_STRIDE) + s01;
#pragma unroll 8
        for (int k = 0; k < HQUADS; ++k) {     // 32 quads: exactly half a chunk per lane
            f32x4 a0 = A0[k], a1 = A1[k], b0 = B0[k], b1 = B1[k];
            acc00 += __builtin_fabsf(a0[0] - b0[0]) + __builtin_fabsf(a0[1] - b0[1])
                   + __builtin_fabsf(a0[2] - b0[2]) + __builtin_fabsf(a0[3] - b0[3]);
            acc01 += __builtin_fabsf(a0[0] - b1[0]) + __builtin_fabsf(a0[1] - b1[1])
                   + __builtin_fabsf(a0[2] - b1[2]) + __builtin_fabsf(a0[3] - b1[3]);
            acc10 += __builtin_fabsf(a1[0] - b0[0]) + __builtin_fabsf(a1[1] - b0[1])
                   + __builtin_fabsf(a1[2] - b0[2]) + __builtin_fabsf(a1[3] - b0[3]);
            acc11 += __builtin_fabsf(a1[0] - b1[0]) + __builtin_fabsf(a1[1] - b1[1])
                   + __builtin_fabsf(a1[2] - b1[2]) + __builtin_fabsf(a1[3] - b1[3]);
        }
        __syncthreads();   // done reading buf[phase] before it is refilled next iteration
        phase ^= 1;
    }

    float* out = partials + (size_t)blockIdx.x * (NIMG * NIMG);
    out[m0 * NIMG + n0] = acc00;
    out[m0 * NIMG + n1] = acc01;
    out[m1 * NIMG + n0] = acc10;
    out[m1 * NIMG + n1] = acc11;
}

// ---------------- finalize: mirror-sum reduction + edge_index -----------------------
__global__ __launch_bounds__(256) void reduce_finalize(
    const float* __restrict__ partials, int* __restrict__ edge_index,
    float* __restrict__ edge_attr) {
    const int p = blockIdx.x * blockDim.x + threadIdx.x;   // 0..1023
    const int i = p >> 5, j = p & (NIMG - 1);
    const int q = j * NIMG + i;                            // mirror slot
    float s = 0.f;
    for (int b = 0; b < KBLOCKS; ++b) {                    // fixed order -> deterministic
        const float* blk = partials + (size_t)b * (NIMG * NIMG);
        s += blk[p] + blk[q];                              // the two complementary K-halves
    }
    edge_attr[p] = (i == j) ? 0.0f : s * (1.0f / (float)D_ELEMS);
    edge_index[p]               = i;                       // repeat(idx, n)
    edge_index[NIMG * NIMG + p] = j;                       // tile(idx, n)
}

// ---------------- host entry --------------------------------------------------------
extern "C" void kernel_launch(void* const* d_in, const int* in_sizes, int n_in,
                              void* d_out, int out_size, void* d_ws, size_t ws_size,
                              hipStream_t stream) {
    const float* images = (const float*)d_in[0];
    float* partials = (float*)d_ws;                        // KBLOCKS * 1024 floats = 1 MB

    pairwise_l1_partials<<<KBLOCKS, 256, 0, stream>>>(images, partials);

    // Output tuple flattened: edge_index [2,1024] int32 (JAX x64 off) then edge_attr [1024]
    int*   ei = (int*)d_out;
    float* ea = (float*)d_out + 2 * NIMG * NIMG;
    reduce_finalize<<<4, 256, 0, stream>>>(partials, ei, ea);
}